// LinearAttention_1580547972190
// MI455X (gfx1250) — compile-verified
//
#include <hip/hip_runtime.h>
#include <hip/hip_bf16.h>
#include <cstddef>

// ---------------------------------------------------------------------------
// LinearAttention on MI455X (gfx1250): fp32 WMMA 16x16x4 for every GEMM,
// async global->LDS DMA (ASYNCcnt) double-buffered staging.
// Memory-bound op (~37 GFLOP vs ~0.65 GB HBM traffic) -> keep fp32.
// ---------------------------------------------------------------------------

typedef __attribute__((ext_vector_type(2))) float v2f;
typedef __attribute__((ext_vector_type(8))) float v8f;
typedef int v4i __attribute__((vector_size(4 * sizeof(int))));

#define B_DIM   16
#define C_DIM   256
#define N_DIM   4096
#define O_QKV   768
#define HEADS   8
#define DHEAD   32
#define KC      16           // K-chunk for the projection GEMMs

#if defined(__gfx1250__) && \
    __has_builtin(__builtin_amdgcn_global_load_async_to_lds_b128) && \
    __has_builtin(__builtin_amdgcn_s_wait_asynccnt)
#define LA_HAVE_ASYNC 1
#endif

__device__ __forceinline__ void la_async_copy16(const float* g, float* l) {
#if defined(LA_HAVE_ASYNC)
  __builtin_amdgcn_global_load_async_to_lds_b128((v4i*)g, (v4i*)l, 0, 0);
#else
  // synchronous fallback, same 16B granule
  l[0] = g[0]; l[1] = g[1]; l[2] = g[2]; l[3] = g[3];
#endif
}

__device__ __forceinline__ v8f wmma_f32_16x16x4(v2f a, v2f b, v8f c) {
#if defined(__gfx1250__) && __has_builtin(__builtin_amdgcn_wmma_f32_16x16x4_f32)
  return __builtin_amdgcn_wmma_f32_16x16x4_f32(false, a, false, b,
                                               (short)0, c, false, false);
#else
  (void)a; (void)b;
  return c;   // host pass only
#endif
}

// ---------------------------------------------------------------------------
// Kernel 1 & 5: Y[b,o,n] = sum_c W[o,c] * X[b,c,n]  (+ optional bias[o])
// WG tile 128(M) x 128(N); 8 waves -> 4x2, each wave 32x64.
// K staged in chunks of 16 via async b128 DMA, 2-stage double buffer.
//   As[buf][mm][20]  (M-major; A frag = one contiguous ds_load_b64)
//   Bs[buf][kk][132] (K-major; B frag = ds_load_2addr_b32 pair)
// ---------------------------------------------------------------------------
__global__ __launch_bounds__(256)
void la_gemm_oc_kernel(const float* __restrict__ W,   // [M_ROWS, 256]
                       const float* __restrict__ X,   // [B, 256, 4096]
                       float* __restrict__ Y,         // [B, M_ROWS, 4096]
                       const float* __restrict__ bias,// [M_ROWS] or nullptr
                       int M_ROWS) {
  __shared__ float As[2][128][20];    // rows 80B (16B aligned), banks ok
  __shared__ float Bs[2][KC][132];    // rows 528B (16B aligned)

  const int b     = blockIdx.z;
  const int oBase = blockIdx.y * 128;
  const int nBase = blockIdx.x * 128;
  const int t     = threadIdx.x;
  const int lane  = t & 31;
  const int wave  = t >> 5;
  const int waveM = (wave & 3) * 32;   // 4 M-subtiles of 32
  const int waveN = (wave >> 2) * 64;  // 2 N-subtiles of 64
  const int lr    = lane & 15;
  const int lh    = lane >> 4;

  const float* __restrict__ Xb = X + (size_t)b * C_DIM * N_DIM;

  v8f acc[2][4];
#pragma unroll
  for (int mi = 0; mi < 2; ++mi)
#pragma unroll
    for (int ni = 0; ni < 4; ++ni)
      acc[mi][ni] = {};

  // ---- stage issuer: 4 x b128 per thread (2 for W tile, 2 for X tile) ----
  auto stage = [&](int buf, int k0) {
#pragma unroll
    for (int i = 0; i < 2; ++i) {
      // W tile: 128 rows x 4 granules of 4 floats = 512 granules
      int c  = t + i * 256;
      int mm = c >> 2;
      int kc = (c & 3) << 2;
      la_async_copy16(&W[(size_t)(oBase + mm) * C_DIM + k0 + kc],
                      &As[buf][mm][kc]);
    }
#pragma unroll
    for (int i = 0; i < 2; ++i) {
      // X tile: 16 rows x 32 granules of 4 floats = 512 granules
      int c  = t + i * 256;
      int kk = c >> 5;
      int nc = (c & 31) << 2;
      la_async_copy16(&Xb[(size_t)(k0 + kk) * N_DIM + nBase + nc],
                      &Bs[buf][kk][nc]);
    }
  };

  constexpr int NC = C_DIM / KC;     // 16 chunks
  stage(0, 0);

  for (int c = 0; c < NC; ++c) {
    const int buf = c & 1;
    const bool has_next = (c + 1 < NC);
    if (has_next) stage(buf ^ 1, (c + 1) * KC);
    if (c + 2 < NC)   // speculative GL2 prefetch two chunks ahead
      __builtin_prefetch(&Xb[(size_t)((c + 2) * KC) * N_DIM + nBase + (t & 127)], 0, 0);

#if defined(LA_HAVE_ASYNC)
    if (has_next) __builtin_amdgcn_s_wait_asynccnt(4);  // retire stage c only
    else          __builtin_amdgcn_s_wait_asynccnt(0);
#endif
    __syncthreads();

#pragma unroll
    for (int ks = 0; ks < KC; ks += 4) {
      v2f afr[2], bfr[4];
#pragma unroll
      for (int mi = 0; mi < 2; ++mi)
        afr[mi] = *(const v2f*)&As[buf][waveM + mi * 16 + lr][ks + 2 * lh];
#pragma unroll
      for (int ni = 0; ni < 4; ++ni) {
        bfr[ni].x = Bs[buf][ks + 2 * lh + 0][waveN + ni * 16 + lr];
        bfr[ni].y = Bs[buf][ks + 2 * lh + 1][waveN + ni * 16 + lr];
      }
#pragma unroll
      for (int mi = 0; mi < 2; ++mi)
#pragma unroll
        for (int ni = 0; ni < 4; ++ni)
          acc[mi][ni] = wmma_f32_16x16x4(afr[mi], bfr[ni], acc[mi][ni]);
    }
    __syncthreads();   // protect buf before stage c+2 overwrites it
  }

  float* __restrict__ Yb = Y + (size_t)b * M_ROWS * N_DIM;
#pragma unroll
  for (int mi = 0; mi < 2; ++mi) {
#pragma unroll
    for (int ni = 0; ni < 4; ++ni) {
      const int col = nBase + waveN + ni * 16 + lr;
#pragma unroll
      for (int r = 0; r < 8; ++r) {
        const int row = oBase + waveM + mi * 16 + r + 8 * lh;
        float v = acc[mi][ni][r];
        if (bias) v += bias[row];
        Yb[(size_t)row * N_DIM + col] = v;
      }
    }
  }
}

// ---------------------------------------------------------------------------
// Kernel 2: per-row softmax stats for K (rows = b*256 + h*32 + d, len 4096)
// ---------------------------------------------------------------------------
__global__ __launch_bounds__(256)
void la_softmax_stats_kernel(const float* __restrict__ qkv,
                             float* __restrict__ smax,
                             float* __restrict__ sinv) {
  __shared__ float red[256];
  const int row = blockIdx.x;
  const int b   = row >> 8;
  const int hd  = row & 255;
  const float* __restrict__ kr =
      qkv + ((size_t)b * O_QKV + 256 + hd) * N_DIM;
  const int t = threadIdx.x;

  float m = -3.4e38f;
  for (int i = t; i < N_DIM; i += 256) m = fmaxf(m, kr[i]);
  red[t] = m;
  __syncthreads();
  for (int s = 128; s > 0; s >>= 1) {
    if (t < s) red[t] = fmaxf(red[t], red[t + s]);
    __syncthreads();
  }
  const float rm = red[0];
  __syncthreads();

  float s = 0.f;
  for (int i = t; i < N_DIM; i += 256) s += __expf(kr[i] - rm);
  red[t] = s;
  __syncthreads();
  for (int s2 = 128; s2 > 0; s2 >>= 1) {
    if (t < s2) red[t] += red[t + s2];
    __syncthreads();
  }
  if (t == 0) {
    smax[row] = rm;
    sinv[row] = 1.f / red[0];
  }
}

// ---------------------------------------------------------------------------
// Kernel 3: ctx[b,h,d,e] = sum_n softmax(K)[d,n] * V[e,n]   (32x32, K=4096)
// One WG per (b,h); K/V chunks staged via async DMA; softmax applied on the
// fly to the A fragment; cross-wave reduce via LDS ds_add_f32.
// ---------------------------------------------------------------------------
__global__ __launch_bounds__(256)
void la_context_kernel(const float* __restrict__ qkv,
                       const float* __restrict__ smax,
                       const float* __restrict__ sinv,
                       float* __restrict__ ctx) {
  __shared__ float Ks[32][132];   // [d][nn], rows 528B (16B aligned)
  __shared__ float Vs[32][132];   // [e][nn]
  __shared__ float Cred[32][33];  // 32x32 reduction buffer (+pad)

  const int bh = blockIdx.x;
  const int b  = bh >> 3;
  const int h  = bh & 7;
  const float* __restrict__ K =
      qkv + ((size_t)b * O_QKV + 256 + h * DHEAD) * N_DIM;
  const float* __restrict__ V =
      qkv + ((size_t)b * O_QKV + 512 + h * DHEAD) * N_DIM;

  const int t    = threadIdx.x;
  const int lane = t & 31;
  const int wave = t >> 5;
  const int lr   = lane & 15;
  const int lh   = lane >> 4;

  const int rowbase = (b << 8) + h * DHEAD;
  const float mx0 = smax[rowbase + lr];
  const float iv0 = sinv[rowbase + lr];
  const float mx1 = smax[rowbase + 16 + lr];
  const float iv1 = sinv[rowbase + 16 + lr];

  v8f acc[2][2];
#pragma unroll
  for (int mi = 0; mi < 2; ++mi)
#pragma unroll
    for (int ni = 0; ni < 2; ++ni)
      acc[mi][ni] = {};

  for (int i = t; i < 32 * 33; i += 256) (&Cred[0][0])[i] = 0.f;

  for (int n0 = 0; n0 < N_DIM; n0 += 128) {
    // stage K and V tiles: 32 rows x 32 granules each -> 4+4 per thread
#pragma unroll
    for (int i = 0; i < 4; ++i) {
      int c  = t + i * 256;
      int d  = c >> 5;
      int nc = (c & 31) << 2;
      la_async_copy16(&K[(size_t)d * N_DIM + n0 + nc], &Ks[d][nc]);
      la_async_copy16(&V[(size_t)d * N_DIM + n0 + nc], &Vs[d][nc]);
    }
#if defined(LA_HAVE_ASYNC)
    __builtin_amdgcn_s_wait_asynccnt(0);
#endif
    __syncthreads();

#pragma unroll
    for (int s = 0; s < 16; s += 4) {
      const int kk = wave * 16 + s;      // this wave's K (=n) slice
      v2f a0, a1, b0, b1;
      a0.x = __expf(Ks[lr][kk + 2 * lh + 0] - mx0) * iv0;
      a0.y = __expf(Ks[lr][kk + 2 * lh + 1] - mx0) * iv0;
      a1.x = __expf(Ks[16 + lr][kk + 2 * lh + 0] - mx1) * iv1;
      a1.y = __expf(Ks[16 + lr][kk + 2 * lh + 1] - mx1) * iv1;
      b0.x = Vs[lr][kk + 2 * lh + 0];
      b0.y = Vs[lr][kk + 2 * lh + 1];
      b1.x = Vs[16 + lr][kk + 2 * lh + 0];
      b1.y = Vs[16 + lr][kk + 2 * lh + 1];
      acc[0][0] = wmma_f32_16x16x4(a0, b0, acc[0][0]);
      acc[0][1] = wmma_f32_16x16x4(a0, b1, acc[0][1]);
      acc[1][0] = wmma_f32_16x16x4(a1, b0, acc[1][0]);
      acc[1][1] = wmma_f32_16x16x4(a1, b1, acc[1][1]);
    }
    __syncthreads();
  }

  // cross-wave reduction (ds_add_f32)
#pragma unroll
  for (int mi = 0; mi < 2; ++mi)
#pragma unroll
    for (int ni = 0; ni < 2; ++ni)
#pragma unroll
      for (int r = 0; r < 8; ++r) {
        int d = mi * 16 + r + 8 * lh;
        int e = ni * 16 + lr;
        atomicAdd(&Cred[d][e], acc[mi][ni][r]);
      }
  __syncthreads();

  float* __restrict__ cout = ctx + (size_t)bh * DHEAD * DHEAD;
  for (int i = t; i < DHEAD * DHEAD; i += 256) {
    int d = i >> 5, e = i & 31;
    cout[i] = Cred[d][e];
  }
}

// ---------------------------------------------------------------------------
// Kernel 4: attn[b, h*32+e, n] = sum_d ctx[d,e] * Q[d,n]    (K=32)
// Grid: (N/128, B*HEADS). Each wave owns a 16-wide n strip, 32 e-rows.
// ---------------------------------------------------------------------------
__global__ __launch_bounds__(256)
void la_apply_q_kernel(const float* __restrict__ qkv,
                       const float* __restrict__ ctx,
                       float* __restrict__ attn) {
  const int bh    = blockIdx.y;
  const int b     = bh >> 3;
  const int h     = bh & 7;
  const int nBase = blockIdx.x * 128;
  const float* __restrict__ Q =
      qkv + ((size_t)b * O_QKV + h * DHEAD) * N_DIM;
  const float* __restrict__ cm = ctx + (size_t)bh * DHEAD * DHEAD; // [d][e]

  const int t    = threadIdx.x;
  const int lane = t & 31;
  const int wave = t >> 5;
  const int lr   = lane & 15;
  const int lh   = lane >> 4;
  const int n0   = nBase + wave * 16;

  v8f acc[2];
  acc[0] = {};
  acc[1] = {};

#pragma unroll
  for (int k = 0; k < DHEAD; k += 4) {
    v2f a0, a1, bf;
    a0.x = cm[(k + 2 * lh + 0) * 32 + lr];
    a0.y = cm[(k + 2 * lh + 1) * 32 + lr];
    a1.x = cm[(k + 2 * lh + 0) * 32 + 16 + lr];
    a1.y = cm[(k + 2 * lh + 1) * 32 + 16 + lr];
    bf.x = Q[(size_t)(k + 2 * lh + 0) * N_DIM + n0 + lr];
    bf.y = Q[(size_t)(k + 2 * lh + 1) * N_DIM + n0 + lr];
    acc[0] = wmma_f32_16x16x4(a0, bf, acc[0]);
    acc[1] = wmma_f32_16x16x4(a1, bf, acc[1]);
  }

  float* __restrict__ outp = attn + ((size_t)b * 256 + h * DHEAD) * N_DIM;
#pragma unroll
  for (int mi = 0; mi < 2; ++mi)
#pragma unroll
    for (int r = 0; r < 8; ++r) {
      int e = mi * 16 + r + 8 * lh;
      outp[(size_t)e * N_DIM + n0 + lr] = acc[mi][r];
    }
}

// ---------------------------------------------------------------------------
// Host-side launch
// ---------------------------------------------------------------------------
extern "C" void kernel_launch(void* const* d_in, const int* in_sizes, int n_in,
                              void* d_out, int out_size, void* d_ws,
                              size_t ws_size, hipStream_t stream) {
  const float* x     = (const float*)d_in[0];   // [16,256,64,64]
  const float* w_qkv = (const float*)d_in[1];   // [768,256]
  const float* w_out = (const float*)d_in[2];   // [256,256]
  const float* b_out = (const float*)d_in[3];   // [256]
  float*       out   = (float*)d_out;           // [16,256,64,64]

  float* ws   = (float*)d_ws;
  float* qkv  = ws;                                   // 16*768*4096
  float* attn = qkv + (size_t)B_DIM * O_QKV * N_DIM;  // 16*256*4096
  float* ctx  = attn + (size_t)B_DIM * 256 * N_DIM;   // 128*32*32
  float* smax = ctx + (size_t)B_DIM * HEADS * DHEAD * DHEAD;
  float* sinv = smax + (size_t)B_DIM * 256;

  // 1) QKV projection: 768x256 * 256x4096 per batch
  dim3 g1(N_DIM / 128, O_QKV / 128, B_DIM);
  la_gemm_oc_kernel<<<g1, 256, 0, stream>>>(w_qkv, x, qkv, nullptr, O_QKV);

  // 2) softmax stats over K rows
  la_softmax_stats_kernel<<<dim3(B_DIM * 256), 256, 0, stream>>>(qkv, smax, sinv);

  // 3) context = softmax(K) @ V^T  per (b,h)
  la_context_kernel<<<dim3(B_DIM * HEADS), 256, 0, stream>>>(qkv, smax, sinv, ctx);

  // 4) attn = ctx^T @ Q  per (b,h)
  la_apply_q_kernel<<<dim3(N_DIM / 128, B_DIM * HEADS), 256, 0, stream>>>(qkv, ctx, attn);

  // 5) output projection (+bias): 256x256 * 256x4096 per batch
  dim3 g5(N_DIM / 128, 256 / 128, B_DIM);
  la_gemm_oc_kernel<<<g5, 256, 0, stream>>>(w_out, attn, out, b_out, 256);
}